// NCCForces3d_69561290326802
// MI455X (gfx1250) — compile-verified
//
#include <hip/hip_runtime.h>
#include <stdint.h>

// NCC forces, fused single pass.
// Volumes: (1,1,256,256,256) f32. Output: (1,3,256,256,256) f32.
#define DW   256      // cube edge
#define TS   32       // tile size in x and y
#define ZCH  64       // z-chunk per workgroup
#define RW   36       // halo-extended tile edge (TS + 4)
#define RSP  33       // padded LDS row stride for row-sums (conflict-free)
#define SLOT (RW * RW)
#define NPIX 125.0f

#if defined(__has_builtin)
#if __has_builtin(__builtin_amdgcn_tensor_load_to_lds)
#define HAVE_TDM 1
#endif
#endif

typedef unsigned int v4u __attribute__((ext_vector_type(4)));
typedef int          v4i __attribute__((ext_vector_type(4)));
typedef int          v8i __attribute__((ext_vector_type(8)));

// ---- CDNA5 async global->LDS (ASYNCcnt path) -------------------------------
__device__ __forceinline__ void async_ld_b32(uint32_t lds_addr, uint32_t voff,
                                             uint64_t sbase) {
  asm volatile("global_load_async_to_lds_b32 %0, %1, %2"
               :: "v"(lds_addr), "v"(voff), "s"(sbase)
               : "memory");
}
__device__ __forceinline__ void wait_async0() {
  asm volatile("s_wait_asynccnt 0x0" ::: "memory");
}

#ifdef HAVE_TDM
// ---- CDNA5 Tensor Data Mover: load a 2D f32 tile into LDS ------------------
// D# per CDNA5 ISA: group0 = {count/type, lds_addr,
// global_addr[56:0], type=2}; group1 = {data_size=4B, tensor dims (remaining
// extents for OOB zero-fill), tile dims, row stride}. 2D => groups 2/3 zero.
__device__ __forceinline__ void tdm_load_slice(uint64_t gaddr, uint32_t lds_addr,
                                               uint32_t dim0_rem, uint32_t dim1_rem,
                                               uint32_t tile_w, uint32_t tile_h) {
  v4u g0;
  g0[0] = 1u;                                             // count=1, user mode
  g0[1] = lds_addr;                                       // LDS byte address
  g0[2] = (uint32_t)gaddr;                                // global_addr[31:0]
  g0[3] = (uint32_t)((gaddr >> 32) & 0x01FFFFFFu) | (2u << 30);  // [56:32]|type=2
  v8i g1;
  g1[0] = (int)(2u << 16);                                // data_size=2 (4B)
  g1[1] = (int)((dim0_rem & 0xFFFFu) << 16);              // tensor_dim0 lo16
  g1[2] = (int)(((dim0_rem >> 16) & 0xFFFFu) |
                ((dim1_rem & 0xFFFFu) << 16));            // dim0 hi | dim1 lo
  g1[3] = (int)(((dim1_rem >> 16) & 0xFFFFu) |
                ((tile_w & 0xFFFFu) << 16));              // dim1 hi | tile_dim0
  g1[4] = (int)(tile_h & 0xFFFFu);                        // tile_dim1, tile_dim2=0
  g1[5] = DW;                                             // tensor_dim0_stride lo
  g1[6] = 0;                                              // stride0 hi | stride1 lo
  g1[7] = 0;
  v4i z4 = {0, 0, 0, 0};
#if defined(__clang_major__) && (__clang_major__ >= 23)
  v8i z8 = {0, 0, 0, 0, 0, 0, 0, 0};
  __builtin_amdgcn_tensor_load_to_lds(g0, g1, z4, z4, z8, 0);
#else
  __builtin_amdgcn_tensor_load_to_lds(g0, g1, z4, z4, 0);
#endif
}
#endif

__global__ __launch_bounds__(1024, 1)
void ncc_forces_kernel(const float* __restrict__ img,
                       const float* __restrict__ ref,
                       float* __restrict__ out) {
  // Raw slice rings (4-deep, both inputs, stride 36 = TDM-contiguous)
  // + x-direction row sums (5 quantities, padded stride 33).
  __shared__ float sI[4 * SLOT];         // 20,736 B
  __shared__ float sR[4 * SLOT];         // 20,736 B
  __shared__ float sRS[5 * RW * RSP];    // 23,760 B

  const int tx = threadIdx.x;            // 0..31 (x within tile)
  const int ty = threadIdx.y;            // 0..31 (y within tile)
  const int t  = ty * 32 + tx;           // 0..1023
  const int bx = blockIdx.x, by = blockIdx.y;
  const int x0 = bx * TS;
  const int y0 = by * TS;
  const int z0 = blockIdx.z * ZCH;
  const int gxo = x0 + tx;               // global output x
  const int gyo = y0 + ty;               // global output y

  const uint64_t imgB = (uint64_t)(uintptr_t)img;
  const uint64_t refB = (uint64_t)(uintptr_t)ref;

  // Register ring of 2D-filtered slices, 5 quantities x 5 depth (oldest first).
  float r00=0,r01=0,r02=0,r03=0,r04=0;   // sum_i
  float r10=0,r11=0,r12=0,r13=0,r14=0;   // sum_r
  float r20=0,r21=0,r22=0,r23=0,r24=0;   // sum_ii
  float r30=0,r31=0,r32=0,r33=0,r34=0;   // sum_rr
  float r40=0,r41=0,r42=0,r43=0,r44=0;   // sum_ir

  for (int s = 0; s < ZCH + 4; ++s) {
    const int zi   = z0 - 2 + s;                 // slice being staged
    const int slot = (s & 3) * SLOT;

    __syncthreads();  // raw-ring slot reuse vs previous emit

    // ---------------- Phase A: stage raw slice zi into LDS ring ----------
    const bool zin = (zi >= 0) && (zi < DW);
    if (!zin) {
      for (int k = t; k < SLOT; k += 1024) {     // OOB z slice: zeros
        sI[slot + k] = 0.0f;
        sR[slot + k] = 0.0f;
      }
    } else {
#ifdef HAVE_TDM
      if (bx != 0) {
        // TDM path: tile start (x0-2, y0-2+ys) is in-bounds; right/bottom
        // overflow zero-filled via tensor-dim OOB; top margin pre-zeroed.
        const int ys = (by == 0) ? 2 : 0;
        if (ys != 0 && t < 2 * RW) {             // rows 0..1 (both arrays)
          sI[slot + t] = 0.0f;
          sR[slot + t] = 0.0f;
        }
        if (ty == 0) {                           // exactly one wave issues TDM
          const int rowstart = y0 - 2 + ys;
          const uint64_t off = 4ull * ((uint64_t)zi * (DW * DW) +
                                       (uint64_t)rowstart * DW + (x0 - 2));
          const uint32_t d0r = (uint32_t)(DW - (x0 - 2));   // remaining x extent
          const uint32_t d1r = (uint32_t)(DW - rowstart);   // remaining y extent
          const uint32_t ldsI = (uint32_t)(uintptr_t)&sI[slot + ys * RW];
          const uint32_t ldsR = (uint32_t)(uintptr_t)&sR[slot + ys * RW];
          tdm_load_slice(imgB + off, ldsI, d0r, d1r, RW, (uint32_t)(RW - ys));
          tdm_load_slice(refB + off, ldsR, d0r, d1r, RW, (uint32_t)(RW - ys));
          __builtin_amdgcn_s_wait_tensorcnt(0);
        }
      } else
#endif
      {
        // Per-lane async path (left-edge tiles / no-TDM fallback).
        for (int k = t; k < SLOT; k += 1024) {
          const int row = k / RW;
          const int col = k - row * RW;
          const int gy  = y0 - 2 + row;
          const int gx  = x0 - 2 + col;
          const int li  = slot + k;
          if ((unsigned)gy < (unsigned)DW && (unsigned)gx < (unsigned)DW) {
            const uint32_t voff = (uint32_t)((gy * DW + gx) * 4);
            const uint64_t sb   = (uint64_t)zi * (uint64_t)(DW * DW * 4);
            async_ld_b32((uint32_t)(uintptr_t)&sI[li], voff, imgB + sb);
            async_ld_b32((uint32_t)(uintptr_t)&sR[li], voff, refB + sb);
          } else {
            sI[li] = 0.0f;   // zero padding, matches jnp.pad semantics
            sR[li] = 0.0f;
          }
        }
        wait_async0();
      }
    }
    __syncthreads();

    // ---------------- Phase B1: x-direction 5-wide row sums --------------
    // 5 quantities * 36 halo rows = 180 sliding-window tasks (branchless).
    if (t < 5 * RW) {
      const int q    = t / RW;
      const int row  = t - q * RW;
      const int soff = slot + row * RW;
      auto val = [&](int col) -> float {
        const float a = sI[soff + col];
        const float b = sR[soff + col];
        float v = a;                       // q==0: i
        v = (q == 1) ? b     : v;          // r
        v = (q == 2) ? a * a : v;          // i*i
        v = (q == 3) ? b * b : v;          // r*r
        v = (q == 4) ? a * b : v;          // i*r
        return v;
      };
      const int ro = q * RW * RSP + row * RSP;
      float c0 = val(0), c1 = val(1), c2 = val(2), c3 = val(3), c4 = val(4);
      float acc = c0 + c1 + c2 + c3 + c4;
      sRS[ro + 0] = acc;
#pragma unroll
      for (int x = 1; x < TS; ++x) {
        const float nv = val(x + 4);
        acc += nv - c0;
        c0 = c1; c1 = c2; c2 = c3; c3 = c4; c4 = nv;
        sRS[ro + x] = acc;
      }
    }
    __syncthreads();

    // ---------------- Phase B2: y-direction sums; advance rings ----------
    {
      const int b = ty * RSP + tx;
#define COLSUM(Q) (sRS[(Q)*RW*RSP + b]           + sRS[(Q)*RW*RSP + b + RSP] + \
                   sRS[(Q)*RW*RSP + b + 2*RSP]   + sRS[(Q)*RW*RSP + b + 3*RSP] + \
                   sRS[(Q)*RW*RSP + b + 4*RSP])
      const float f0 = COLSUM(0);
      const float f1 = COLSUM(1);
      const float f2 = COLSUM(2);
      const float f3 = COLSUM(3);
      const float f4 = COLSUM(4);
#undef COLSUM
      r00=r01; r01=r02; r02=r03; r03=r04; r04=f0;
      r10=r11; r11=r12; r12=r13; r13=r14; r14=f1;
      r20=r21; r21=r22; r22=r23; r23=r24; r24=f2;
      r30=r31; r31=r32; r32=r33; r33=r34; r34=f3;
      r40=r41; r41=r42; r42=r43; r43=r44; r44=f4;
    }

    // ---------------- Phase C: emit output slice z = zi - 2 --------------
    if (s >= 4) {
      const int z = zi - 2;
      const float S_i  = r00 + r01 + r02 + r03 + r04;
      const float S_r  = r10 + r11 + r12 + r13 + r14;
      const float S_ii = r20 + r21 + r22 + r23 + r24;
      const float S_rr = r30 + r31 + r32 + r33 + r34;
      const float S_ir = r40 + r41 + r42 + r43 + r44;

      const int cur = ((s - 2) & 3) * SLOT;
      const int prv = ((s - 3) & 3) * SLOT;
      const int nxt = ((s - 1) & 3) * SLOT;
      const int ic  = (ty + 2) * RW + (tx + 2);

      const float iv = sI[cur + ic];
      const float rv = sR[cur + ic];
      const float aC  = 0.5f * (iv + rv);               // symmetric avg
      const float aZp = 0.5f * (sI[nxt + ic] + sR[nxt + ic]);
      const float aZm = 0.5f * (sI[prv + ic] + sR[prv + ic]);
      const float aYp = 0.5f * (sI[cur + ic + RW] + sR[cur + ic + RW]);
      const float aYm = 0.5f * (sI[cur + ic - RW] + sR[cur + ic - RW]);
      const float aXp = 0.5f * (sI[cur + ic + 1] + sR[cur + ic + 1]);
      const float aXm = 0.5f * (sI[cur + ic - 1] + sR[cur + ic - 1]);

      // torch.gradient: central inside, one-sided at edges.
      const float gdz = (z   == 0)      ? (aZp - aC)
                      : (z   == DW - 1) ? (aC - aZm) : 0.5f * (aZp - aZm);
      const float gdy = (gyo == 0)      ? (aYp - aC)
                      : (gyo == DW - 1) ? (aC - aYm) : 0.5f * (aYp - aYm);
      const float gdx = (gxo == 0)      ? (aXp - aC)
                      : (gxo == DW - 1) ? (aC - aXm) : 0.5f * (aXp - aXm);

      // Faithful to reference: both "means" use sum_i.
      const float m     = S_i / NPIX;
      const float var_r = S_rr - 2.0f * m * S_r + NPIX * m * m;
      const float var_i = S_ii - 2.0f * m * S_i + NPIX * m * m;
      const float cross = S_ir - m * S_i - m * S_r + NPIX * m * m;
      const float factor = 2.0f * cross / (var_i * var_r + 1e-6f) *
                           (iv - (cross / var_r) * rv);
      const float o = -factor;

      const size_t N   = (size_t)DW * DW * DW;
      const size_t idx = ((size_t)z * DW + gyo) * DW + gxo;
      out[idx]         = o * gdz;   // channel 0: axis 2 (D/z)
      out[N + idx]     = o * gdy;   // channel 1: axis 3 (H/y)
      out[2 * N + idx] = o * gdx;   // channel 2: axis 4 (W/x)
    }
  }
}

extern "C" void kernel_launch(void* const* d_in, const int* in_sizes, int n_in,
                              void* d_out, int out_size, void* d_ws, size_t ws_size,
                              hipStream_t stream) {
  (void)in_sizes; (void)n_in; (void)out_size; (void)d_ws; (void)ws_size;
  // setup_inputs order: image, mask, reference_image, reference_mask
  const float* img = (const float*)d_in[0];
  const float* ref = (const float*)d_in[2];
  float* out = (float*)d_out;

  dim3 grid(DW / TS, DW / TS, DW / ZCH);   // 8 x 8 x 4 = 256 workgroups
  dim3 block(32, 32, 1);                   // 1024 threads = 32 waves (wave32)
  ncc_forces_kernel<<<grid, block, 0, stream>>>(img, ref, out);
}